// MeshUnpool_20615843021401
// MI455X (gfx1250) — compile-verified
//
#include <hip/hip_runtime.h>

typedef __attribute__((ext_vector_type(2))) float v2f;
typedef __attribute__((ext_vector_type(8))) float v8f;

// Problem constants (reference: B, NF, E, T = 4, 64, 4000, 6000)
constexpr int BB = 4;
constexpr int NF = 64;
constexpr int EE = 4000;
constexpr int TT = 6000;

constexpr int KC     = 80;            // K-chunk staged in LDS
constexpr int NCHUNK = EE / KC;       // 50
constexpr int MT     = NF / 16;       // 4 M-tiles per wave (full M)
constexpr int LROW   = KC + 2;        // padded LDS row (82) -> no stride-64 bank conflicts
constexpr int NTILES = TT / 16;       // 375 N-tiles
constexpr int WAVES  = 8;             // waves per block
constexpr int NBX    = (NTILES + WAVES - 1) / WAVES; // 47
constexpr int NLD    = NF * (KC / 4) / 256;          // float4 loads per thread per chunk = 5

__global__ __launch_bounds__(256)
void meshunpool_wmma_f32(const float* __restrict__ features,
                         const float* __restrict__ group,
                         const float* __restrict__ occ,
                         float* __restrict__ out)
{
    __shared__ float sA[2][NF * LROW];   // double-buffered 64 x 82 fp32 = 2 x 20.5 KB

    const int b    = blockIdx.y;
    const int wave = threadIdx.x >> 5;
    const int lane = threadIdx.x & 31;
    const int ln   = lane & 15;                 // N (and M) sub-index
    const int ko   = (lane < 16) ? 0 : 2;       // K-pair offset per WMMA layout

    const int  ntile  = blockIdx.x * WAVES + wave;
    const bool active = (ntile < NTILES);       // wave-uniform -> EXEC all-1s for WMMA
    const int  t0     = ntile * 16;

    const float* fb = features + (size_t)b * NF * EE;
    const float* gb = group    + (size_t)b * EE * TT;
    const float* gcol = gb + (active ? (t0 + ln) : 0);  // + row*TT added per K-step

    // Precompute staging address components (constant across chunks)
    int goff[NLD], loff[NLD];
#pragma unroll
    for (int j = 0; j < NLD; ++j) {
        const int idx = threadIdx.x + j * 256;
        const int row = idx / (KC / 4);
        const int c4  = idx % (KC / 4);
        goff[j] = row * EE + c4 * 4;       // element offset into fb
        loff[j] = row * LROW + c4 * 4;     // element offset into sA buffer
    }

    v8f c[MT];
#pragma unroll
    for (int m = 0; m < MT; ++m) c[m] = (v8f)0.0f;

    float4 pre[NLD];

    // Prologue: load + store chunk 0 into buffer 0
#pragma unroll
    for (int j = 0; j < NLD; ++j)
        pre[j] = *(const float4*)(fb + goff[j]);
#pragma unroll
    for (int j = 0; j < NLD; ++j) {
        float* d = &sA[0][loff[j]];
        *(v2f*)(d)     = v2f{pre[j].x, pre[j].y};   // 8B-aligned -> ds_store_b64
        *(v2f*)(d + 2) = v2f{pre[j].z, pre[j].w};
    }

    for (int ch = 0; ch < NCHUNK; ++ch) {
        __syncthreads();   // buffer (ch&1) now valid for all waves

        // Issue next chunk's global loads; latency overlaps the compute below.
        if (ch + 1 < NCHUNK) {
            const int e1 = (ch + 1) * KC;
#pragma unroll
            for (int j = 0; j < NLD; ++j)
                pre[j] = *(const float4*)(fb + goff[j] + e1);
        }

        if (active) {
            const int e0 = ch * KC;
            const float* sbuf = sA[ch & 1];
#pragma unroll 4
            for (int kk = 0; kk < KC; kk += 4) {
                // B fragment: lane holds {group[e+ko][t], group[e+ko+1][t]}
                // group is streamed exactly once -> non-temporal to spare L2 for A
                const size_t r = (size_t)(e0 + kk + ko) * TT;
                v2f Bf;
                Bf.x = __builtin_nontemporal_load(gcol + r);
                Bf.y = __builtin_nontemporal_load(gcol + r + TT);
#pragma unroll
                for (int m = 0; m < MT; ++m) {
                    // A fragment: lane holds {A[M][kk+ko], A[M][kk+ko+1]} from LDS
                    const v2f Af = *(const v2f*)&sbuf[(m * 16 + ln) * LROW + kk + ko];
                    c[m] = __builtin_amdgcn_wmma_f32_16x16x4_f32(
                        /*neg_a=*/false, Af, /*neg_b=*/false, Bf,
                        /*c_mod=*/(short)0, c[m],
                        /*reuse_a=*/false, /*reuse_b=*/false);
                }
            }
        }

        // Store prefetched chunk into the other buffer. Safe with one barrier:
        // every wave passed the top-of-loop barrier, so no wave is still
        // reading buffer ((ch+1)&1) from iteration ch-1.
        if (ch + 1 < NCHUNK) {
            float* dbuf = sA[(ch + 1) & 1];
#pragma unroll
            for (int j = 0; j < NLD; ++j) {
                float* d = &dbuf[loff[j]];
                *(v2f*)(d)     = v2f{pre[j].x, pre[j].y};
                *(v2f*)(d + 2) = v2f{pre[j].z, pre[j].w};
            }
        }
    }

    if (active) {
        // fold occurrence normalization into the output scale (column-wise)
        const float rocc = 1.0f / occ[(size_t)b * TT + t0 + ln];
        const int hi = (lane < 16) ? 0 : 8;
#pragma unroll
        for (int m = 0; m < MT; ++m) {
#pragma unroll
            for (int vg = 0; vg < 8; ++vg) {
                const int f = m * 16 + vg + hi;   // C layout: VGPR vg -> row vg (+8 for hi lanes)
                out[(size_t)(b * NF + f) * TT + t0 + ln] = c[m][vg] * rocc;
            }
        }
    }
}

extern "C" void kernel_launch(void* const* d_in, const int* in_sizes, int n_in,
                              void* d_out, int out_size, void* d_ws, size_t ws_size,
                              hipStream_t stream) {
    (void)in_sizes; (void)n_in; (void)out_size; (void)d_ws; (void)ws_size;
    const float* features = (const float*)d_in[0];
    const float* group    = (const float*)d_in[1];
    const float* occ      = (const float*)d_in[2];
    float* out            = (float*)d_out;

    dim3 grid(NBX, BB, 1);
    dim3 block(256, 1, 1);
    meshunpool_wmma_f32<<<grid, block, 0, stream>>>(features, group, occ, out);
}